// LargeScaleSourceIntegration_4440996184190
// MI455X (gfx1250) — compile-verified
//
#include <hip/hip_runtime.h>
#include <hip/hip_bf16.h>
#include <cstdint>

// ---------------------------------------------------------------------------
// Problem dims (fixed by reference): B=256, D=1024, H=512, E=32, K=4, C=32
// ---------------------------------------------------------------------------
#define BB 256
#define DD 1024
#define HH 512
#define EE 32
#define KK 4
#define CC 32

typedef float v2f __attribute__((ext_vector_type(2)));
typedef float v8f __attribute__((ext_vector_type(8)));
typedef int   v4i __attribute__((ext_vector_type(4)));

typedef __attribute__((address_space(1))) v4i* as1_v4i_ptr;
typedef __attribute__((address_space(3))) v4i* as3_v4i_ptr;

#if __has_builtin(__builtin_amdgcn_global_load_async_to_lds_b128)
#define USE_ASYNC_A 1
#else
#define USE_ASYNC_A 0
#endif

#if __has_builtin(__builtin_amdgcn_s_wait_asynccnt)
#define WAIT_ASYNC0() __builtin_amdgcn_s_wait_asynccnt(0)
#else
#define WAIT_ASYNC0() asm volatile("s_wait_asynccnt 0" ::: "memory")
#endif

// ---------------------------------------------------------------------------
// Batched GEMM: Out[e] = act(A[e] (MxK) * W[e] (KxN) + bias[e])
//
// Block: 128 threads = 4 waves. Block tile 128(M) x 64(N). K-chunk = 32,
// double-buffered LDS (sA 16KB x2, sB 8KB x2). A tile staged via
// GLOBAL_LOAD_ASYNC_TO_LDS_B128 (ASYNCcnt path) when available, B tile via
// register interleave; copies for chunk i+1 overlap the 64 WMMAs of chunk i.
//
// Each wave owns a 32(M) x 64(N) strip: 2 A-frags x 4 B-frags -> 8
// accumulators of v8f via V_WMMA_F32_16X16X4_F32.
//
// LDS layouts:
//   sA: row-major [128][32]            -> A frag = one ds_load_b64
//   sB: k-pair interleaved [16][64][2] -> B frag = one ds_load_b64 (no movs)
//
// Pipeline sync per chunk: each wave waits its own ASYNCcnt to 0, then the
// workgroup barrier publishes all async writes before any wave reads them.
// ---------------------------------------------------------------------------
template <int RELU>
__global__ __launch_bounds__(128)
void gemm_wmma_f32(const float* __restrict__ A, const float* __restrict__ W,
                   const float* __restrict__ bias, float* __restrict__ Out,
                   int M, int Kdim, int N,
                   long strideA, long strideW, long strideBias, long strideOut)
{
    const int e    = blockIdx.z;
    const int tid  = threadIdx.x;
    const int wave = tid >> 5;
    const int lane = tid & 31;
    const int bm   = blockIdx.y * 128;
    const int bn   = blockIdx.x * 64;

    const float* Ae = A + (long)e * strideA;
    const float* We = W + (long)e * strideW;
    const float* be = bias + (long)e * strideBias;
    float*       Oe = Out + (long)e * strideOut;

    __shared__ __align__(16) float sA[2][128 * 32];
    __shared__ __align__(16) float sB[2][32 * 64];

    v8f acc[2][4];
#pragma unroll
    for (int fm = 0; fm < 2; ++fm)
#pragma unroll
        for (int nf = 0; nf < 4; ++nf) acc[fm][nf] = (v8f)0.0f;

    const int lo = lane & 15;         // row-in-frag / n-in-frag
    const int hi = (lane >> 4) << 1;  // 0 or 2: K sub-offset for this half-wave

    // B staging assignment: thread -> (k-row pair, 8-column segment)
    const int pr    = tid >> 3;        // 0..15
    const int cbase = (tid & 7) << 3;  // 0,8,..,56

    float4 pb[4];

#if !USE_ASYNC_A
    float4 pa[8];
#endif

    // ---- A tile staging: global [128][Kdim] chunk -> sA[buf] [128][32] ----
#if USE_ASYNC_A
    auto loadA = [&](int buf, int kc) {
#pragma unroll
        for (int i = 0; i < 8; ++i) {
            int idx = i * 128 + tid;          // 0..1023 float4s, coalesced
            int r = idx >> 3;                 // 8 float4 per 32-float row
            int c = (idx & 7) << 2;
            const float* src = &Ae[(long)(bm + r) * Kdim + kc + c];
            const float* dst = &sA[buf][r * 32 + c];
            __builtin_amdgcn_global_load_async_to_lds_b128(
                (as1_v4i_ptr)(unsigned long long)(uintptr_t)src,
                (as3_v4i_ptr)(unsigned int)(uintptr_t)dst,
                0, 0);
        }
    };
#else
    auto loadA = [&](int kc) {
#pragma unroll
        for (int i = 0; i < 8; ++i) {
            int idx = i * 128 + tid;
            int r = idx >> 3;
            int c = (idx & 7) << 2;
            pa[i] = *(const float4*)&Ae[(long)(bm + r) * Kdim + kc + c];
        }
    };
    auto stageA = [&](int buf) {
#pragma unroll
        for (int i = 0; i < 8; ++i) {
            int idx = i * 128 + tid;
            int r = idx >> 3;
            int c = (idx & 7) << 2;
            *(float4*)&sA[buf][r * 32 + c] = pa[i];
        }
    };
#endif

    auto loadB = [&](int kc) {
        const float4 r0a = *(const float4*)&We[(long)(kc + 2 * pr)     * N + bn + cbase];
        const float4 r0b = *(const float4*)&We[(long)(kc + 2 * pr)     * N + bn + cbase + 4];
        const float4 r1a = *(const float4*)&We[(long)(kc + 2 * pr + 1) * N + bn + cbase];
        const float4 r1b = *(const float4*)&We[(long)(kc + 2 * pr + 1) * N + bn + cbase + 4];
        pb[0] = make_float4(r0a.x, r1a.x, r0a.y, r1a.y);
        pb[1] = make_float4(r0a.z, r1a.z, r0a.w, r1a.w);
        pb[2] = make_float4(r0b.x, r1b.x, r0b.y, r1b.y);
        pb[3] = make_float4(r0b.z, r1b.z, r0b.w, r1b.w);
    };
    auto stageB = [&](int buf) {
        const int base = pr * 128 + cbase * 2;
#pragma unroll
        for (int i = 0; i < 4; ++i)
            *(float4*)&sB[buf][base + 4 * i] = pb[i];
    };

    const int nch = Kdim >> 5;
#if USE_ASYNC_A
    loadA(0, 0);
#else
    loadA(0);
#endif
    loadB(0);

    for (int ch = 0; ch < nch; ++ch) {
        const int buf = ch & 1;
        stageB(buf);
#if USE_ASYNC_A
        WAIT_ASYNC0();                 // my async copies into sA[buf] done
#else
        stageA(buf);
#endif
        __syncthreads();               // publish sA[buf] + sB[buf] to all waves
        if (ch + 1 < nch) {            // overlaps WMMA below
#if USE_ASYNC_A
            loadA(buf ^ 1, (ch + 1) << 5);
#else
            loadA((ch + 1) << 5);
#endif
            loadB((ch + 1) << 5);
        }

#pragma unroll
        for (int kk = 0; kk < 8; ++kk) {          // 8 k-steps of 4
            const int kb = kk * 4 + hi;           // even
            v2f a0 = *(const v2f*)&sA[buf][(wave * 32 + lo)      * 32 + kb];
            v2f a1 = *(const v2f*)&sA[buf][(wave * 32 + 16 + lo) * 32 + kb];
#pragma unroll
            for (int nf = 0; nf < 4; ++nf) {
                v2f b = *(const v2f*)&sB[buf][(kb >> 1) * 128 + (nf * 16 + lo) * 2];
                acc[0][nf] = __builtin_amdgcn_wmma_f32_16x16x4_f32(
                    false, a0, false, b, (short)0, acc[0][nf], false, false);
                acc[1][nf] = __builtin_amdgcn_wmma_f32_16x16x4_f32(
                    false, a1, false, b, (short)0, acc[1][nf], false, false);
            }
        }
    }

    // ---- epilogue: bias + optional ReLU ----
#pragma unroll
    for (int fm = 0; fm < 2; ++fm) {
        const int rbase = bm + wave * 32 + fm * 16 + ((lane >> 4) << 3);
#pragma unroll
        for (int nf = 0; nf < 4; ++nf) {
            const int col = bn + nf * 16 + lo;
            const float bv = be[col];
#pragma unroll
            for (int j = 0; j < 8; ++j) {
                float v = acc[fm][nf][j] + bv;
                if (RELU) v = fmaxf(v, 0.0f);
                Oe[(long)(rbase + j) * N + col] = v;
            }
        }
    }
}

// ---------------------------------------------------------------------------
// Gating tail: logits = g @ Wg2 + bg2, softmax, sparsity count, top-4.
// One wave (32 lanes == E) per token.
// ---------------------------------------------------------------------------
__global__ __launch_bounds__(32)
void gating_kernel(const float* __restrict__ g, const float* __restrict__ Wg2,
                   const float* __restrict__ bg2,
                   float* __restrict__ weights_out, int* __restrict__ topk_idx,
                   float* __restrict__ spars_cnt)
{
    const int b = blockIdx.x;
    const int e = threadIdx.x;

    const float* gr = g + (long)b * HH;
    float acc = bg2[e];
    for (int k = 0; k < HH; ++k)
        acc = fmaf(gr[k], Wg2[k * EE + e], acc);

    // softmax across the wave
    float m = acc;
    for (int off = 16; off; off >>= 1)
        m = fmaxf(m, __shfl_xor(m, off, 32));
    float ex = __expf(acc - m);
    float s = ex;
    for (int off = 16; off; off >>= 1)
        s += __shfl_xor(s, off, 32);
    const float w = ex / s;
    weights_out[b * EE + e] = w;

    unsigned long long ball = __ballot(w > 0.01f);
    if (e == 0) spars_cnt[b] = (float)__popcll(ball);

    // iterative top-4 argmax (ties -> lowest index, matching lax.top_k)
    float cur = w;
#pragma unroll
    for (int j = 0; j < KK; ++j) {
        float bv = cur;
        int   bi = e;
        for (int off = 16; off; off >>= 1) {
            float ov = __shfl_xor(bv, off, 32);
            int   oi = __shfl_xor(bi, off, 32);
            if (ov > bv || (ov == bv && oi < bi)) { bv = ov; bi = oi; }
        }
        if (e == 0) topk_idx[b * KK + j] = bi;
        if (e == bi) cur = -1.0f;
    }
}

// ---------------------------------------------------------------------------
// Confidence head: one wave per (e,b). Lane c handles confidence channel c.
// conf_be[b,e] = sigmoid( sum_c relu(y[e,b,:]@Wc1[e,:,c] + bc1) * Wc2[e,c] + bc2 )
// ---------------------------------------------------------------------------
__global__ __launch_bounds__(32)
void conf_kernel(const float* __restrict__ y, const float* __restrict__ Wc1,
                 const float* __restrict__ bc1, const float* __restrict__ Wc2,
                 const float* __restrict__ bc2, float* __restrict__ conf_be)
{
    const int idx = blockIdx.x;       // e * B + b
    const int e = idx >> 8;           // B == 256
    const int b = idx & 255;
    const int c = threadIdx.x;

    const float* yr = y + ((long)e * BB + b) * DD;
    const float* wc = Wc1 + (long)e * DD * CC;

    float acc = bc1[e * CC + c];
    for (int k = 0; k < DD; ++k)
        acc = fmaf(yr[k], wc[k * CC + c], acc);

    float h = fmaxf(acc, 0.0f) * Wc2[e * CC + c];
    for (int off = 16; off; off >>= 1)
        h += __shfl_xor(h, off, 32);
    if (c == 0) {
        float z = h + bc2[e];
        conf_be[b * EE + e] = 1.0f / (1.0f + __expf(-z));
    }
}

// ---------------------------------------------------------------------------
// Combine prep: per token, cw = mask*weights*conf, normalize by (sum + 1e-8),
// emit conf_sel in top-k order.
// ---------------------------------------------------------------------------
__global__ __launch_bounds__(32)
void combine_prep(const float* __restrict__ weights,
                  const float* __restrict__ conf_be,
                  const int* __restrict__ topk_idx,
                  float* __restrict__ cwn, float* __restrict__ conf_sel)
{
    const int b = blockIdx.x;
    const int e = threadIdx.x;

    const int t0 = topk_idx[b * KK + 0];
    const int t1 = topk_idx[b * KK + 1];
    const int t2 = topk_idx[b * KK + 2];
    const int t3 = topk_idx[b * KK + 3];
    const bool sel = (e == t0) | (e == t1) | (e == t2) | (e == t3);

    float cw = sel ? weights[b * EE + e] * conf_be[b * EE + e] : 0.0f;
    float denom = cw;
    for (int off = 16; off; off >>= 1)
        denom += __shfl_xor(denom, off, 32);
    denom += 1e-8f;
    cwn[b * EE + e] = cw / denom;

    if (e < KK)
        conf_sel[b * KK + e] = conf_be[b * EE + topk_idx[b * KK + e]];
}

// ---------------------------------------------------------------------------
// Combine: out[b,:] = sum_{j<4} cwn[b, idx_j] * y[idx_j, b, :]
// One block per token; 256 threads x float4 covers D=1024.
// ---------------------------------------------------------------------------
__global__ __launch_bounds__(256)
void combine_kernel(const float* __restrict__ y, const float* __restrict__ cwn,
                    const int* __restrict__ topk_idx, float* __restrict__ out)
{
    const int b = blockIdx.x;
    const int t = threadIdx.x;

    float4 accv = make_float4(0.f, 0.f, 0.f, 0.f);
#pragma unroll
    for (int j = 0; j < KK; ++j) {
        const int e = topk_idx[b * KK + j];
        const float w = cwn[b * EE + e];
        const float4 yv = *(const float4*)&y[((long)e * BB + b) * DD + t * 4];
        accv.x = fmaf(w, yv.x, accv.x);
        accv.y = fmaf(w, yv.y, accv.y);
        accv.z = fmaf(w, yv.z, accv.z);
        accv.w = fmaf(w, yv.w, accv.w);
    }
    *(float4*)&out[(long)b * DD + t * 4] = accv;
}

// ---------------------------------------------------------------------------
// Sparsity finalize: mean over B*E of (weights > 0.01), from per-token counts.
// ---------------------------------------------------------------------------
__global__ __launch_bounds__(32)
void sparsity_kernel(const float* __restrict__ spars_cnt, float* __restrict__ out)
{
    const int t = threadIdx.x;
    float s = 0.0f;
#pragma unroll
    for (int i = 0; i < 8; ++i)
        s += spars_cnt[t + 32 * i];
    for (int off = 16; off; off >>= 1)
        s += __shfl_xor(s, off, 32);
    if (t == 0) out[0] = s / (float)(BB * EE);
}

// ---------------------------------------------------------------------------
// Host-side launch
// ---------------------------------------------------------------------------
extern "C" void kernel_launch(void* const* d_in, const int* in_sizes, int n_in,
                              void* d_out, int out_size, void* d_ws, size_t ws_size,
                              hipStream_t stream)
{
    (void)in_sizes; (void)n_in; (void)out_size; (void)ws_size;

    const float* x   = (const float*)d_in[0];
    const float* Wg1 = (const float*)d_in[1];
    const float* bg1 = (const float*)d_in[2];
    const float* Wg2 = (const float*)d_in[3];
    const float* bg2 = (const float*)d_in[4];
    const float* W1  = (const float*)d_in[5];
    const float* b1  = (const float*)d_in[6];
    const float* W2  = (const float*)d_in[7];
    const float* b2  = (const float*)d_in[8];
    const float* W3  = (const float*)d_in[9];
    const float* b3  = (const float*)d_in[10];
    const float* Wc1 = (const float*)d_in[11];
    const float* bc1 = (const float*)d_in[12];
    const float* Wc2 = (const float*)d_in[13];
    const float* bc2 = (const float*)d_in[14];

    float* out_main  = (float*)d_out;                 // [B, D]
    float* out_w     = out_main + BB * DD;            // [B, E]
    float* out_csel  = out_w + BB * EE;               // [B, K]
    float* out_spars = out_csel + BB * KK;            // [1]

    float* ws      = (float*)d_ws;
    float* g       = ws;                              // B*H
    float* h1      = g + (long)BB * HH;               // E*B*H
    float* h2      = h1 + (long)EE * BB * HH;         // E*B*H
    float* y       = h2 + (long)EE * BB * HH;         // E*B*D
    float* conf_be = y + (long)EE * BB * DD;          // B*E
    float* cwn     = conf_be + (long)BB * EE;         // B*E
    float* scnt    = cwn + (long)BB * EE;             // B
    int*   topk    = (int*)(scnt + BB);               // B*K

    // 1) gating hidden: g = relu(x @ Wg1 + bg1)   [256,1024]x[1024,512]
    gemm_wmma_f32<1><<<dim3(HH / 64, BB / 128, 1), 128, 0, stream>>>(
        x, Wg1, bg1, g, BB, DD, HH, 0, 0, 0, 0);

    // 2) logits/softmax/top-k/sparsity counts
    gating_kernel<<<BB, 32, 0, stream>>>(g, Wg2, bg2, out_w, topk, scnt);

    // 3) expert layer 1: h1[e] = relu(x @ W1[e] + b1[e])
    gemm_wmma_f32<1><<<dim3(HH / 64, BB / 128, EE), 128, 0, stream>>>(
        x, W1, b1, h1, BB, DD, HH,
        0, (long)DD * HH, HH, (long)BB * HH);

    // 4) expert layer 2: h2[e] = relu(h1[e] @ W2[e] + b2[e])
    gemm_wmma_f32<1><<<dim3(HH / 64, BB / 128, EE), 128, 0, stream>>>(
        h1, W2, b2, h2, BB, HH, HH,
        (long)BB * HH, (long)HH * HH, HH, (long)BB * HH);

    // 5) expert layer 3: y[e] = h2[e] @ W3[e] + b3[e]
    gemm_wmma_f32<0><<<dim3(DD / 64, BB / 128, EE), 128, 0, stream>>>(
        h2, W3, b3, y, BB, HH, DD,
        (long)BB * HH, (long)HH * DD, DD, (long)BB * DD);

    // 6) confidence per (expert, token)
    conf_kernel<<<EE * BB, 32, 0, stream>>>(y, Wc1, bc1, Wc2, bc2, conf_be);

    // 7) combine weights + conf_sel
    combine_prep<<<BB, 32, 0, stream>>>(out_w, conf_be, topk, cwn, out_csel);

    // 8) weighted combine over selected experts
    combine_kernel<<<BB, 256, 0, stream>>>(y, cwn, topk, out_main);

    // 9) sparsity scalar
    sparsity_kernel<<<1, 32, 0, stream>>>(scnt, out_spars);
}